// RNNLM_Model_58755152609453
// MI455X (gfx1250) — compile-verified
//
#include <hip/hip_runtime.h>
#include <hip/hip_bf16.h>
#include <stdint.h>

#define VOCAB 50257
#define EMBED 256
#define HID   256
#define STEPS 64
#define BATCH 64

typedef __attribute__((ext_vector_type(2))) float v2f;
typedef __attribute__((ext_vector_type(8))) float v8f;

// D = A(16x4 f32) * B(4x16 f32) + C(16x16 f32)  -- CDNA5 fp32 matrix path
__device__ __forceinline__ v8f wmma_f32(v2f a, v2f b, v8f c) {
    return __builtin_amdgcn_wmma_f32_16x16x4_f32(
        /*neg_a=*/false, a, /*neg_b=*/false, b,
        /*c_mod=*/(short)0, c, /*reuse_a=*/false, /*reuse_b=*/false);
}

// ---------------------------------------------------------------------------
// Phase 1: xI[t][b][:] = emb[input_x[b][t]] @ I      (gathered GEMM, fp32 WMMA)
// grid (64, 4), block 256 (8 waves): block tile 64(M) x 64(N),
// wave tile 16(M) x 32(N) -> 2 accumulators.
// ---------------------------------------------------------------------------
__global__ __launch_bounds__(256) void rnn_xproj(
    const int*   __restrict__ idx,   // [BATCH][STEPS]
    const float* __restrict__ emb,   // [VOCAB][EMBED]
    const float* __restrict__ I,     // [EMBED][HID]
    float*       __restrict__ xI)    // [STEPS*BATCH][HID]
{
    const int lane  = threadIdx.x & 31;
    const int wid   = threadIdx.x >> 5;
    const int m0    = blockIdx.x * 64 + (wid >> 1) * 16;
    const int n0    = blockIdx.y * 64 + (wid & 1) * 32;
    const int l16   = lane & 15;
    const int khalf = (lane >> 4) * 2;          // 0 or 2 (fp32 A layout)

    const int r = m0 + l16;                      // row r = t*BATCH + b
    const int t = r >> 6;
    const int b = r & 63;
    const int token = idx[b * STEPS + t];
    const float* arow = emb + (long)token * EMBED + khalf;

    v8f acc0 = {}; v8f acc1 = {};
    for (int k = 0; k < EMBED; k += 4) {
        v2f a = *(const v2f*)(arow + k);
        const float* bp = I + (long)(k + khalf) * HID + n0 + l16;
        v2f b0, b1;
        b0.x = bp[0];     b0.y = bp[HID];
        b1.x = bp[16];    b1.y = bp[HID + 16];
        acc0 = wmma_f32(a, b0, acc0);
        acc1 = wmma_f32(a, b1, acc1);
    }
    const int rowbase = m0 + (lane >> 4) * 8;
    float* outp = xI + (long)rowbase * HID + n0 + l16;
    #pragma unroll
    for (int j = 0; j < 8; ++j) {
        outp[(long)j * HID]      = acc0[j];
        outp[(long)j * HID + 16] = acc1[j];
    }
}

// ---------------------------------------------------------------------------
// Phase 2: sequential recurrence, single workgroup of 512 threads (16 waves).
// h (64x256 fp32 = 64 KB) lives in LDS; H (256 KB) streams from L2.
// ---------------------------------------------------------------------------
__global__ __launch_bounds__(512) void rnn_recur(
    const float* __restrict__ xI,    // [STEPS*BATCH][HID]
    const float* __restrict__ Hm,    // [HID][HID]
    const float* __restrict__ b1,    // [HID]
    const float* __restrict__ h0,    // [BATCH][HID]
    float*       __restrict__ hs,    // [STEPS*BATCH][HID]
    float*       __restrict__ fin)   // [BATCH][HID]
{
    __shared__ float hbuf[BATCH * HID];
    for (int i = threadIdx.x; i < BATCH * HID; i += 512) hbuf[i] = h0[i];
    __syncthreads();

    const int lane  = threadIdx.x & 31;
    const int wid   = threadIdx.x >> 5;          // 0..15
    const int m0    = (wid >> 2) * 16;
    const int n0    = (wid & 3) * 64;
    const int l16   = lane & 15;
    const int khalf = (lane >> 4) * 2;
    const int rowbase = m0 + (lane >> 4) * 8;

    for (int t = 0; t < STEPS; ++t) {
        const float* xt = xI + (long)t * BATCH * HID;
        v8f acc[4];
        #pragma unroll
        for (int j = 0; j < 4; ++j) {
            const int col = n0 + j * 16 + l16;
            const float bb = b1[col];
            #pragma unroll
            for (int rr = 0; rr < 8; ++rr)
                acc[j][rr] = xt[(long)(rowbase + rr) * HID + col] + bb;
        }
        const float* arow = hbuf + (m0 + l16) * HID + khalf;
        for (int k = 0; k < HID; k += 4) {
            v2f a = *(const v2f*)(arow + k);                   // ds_load_b64
            const float* br0 = Hm + (long)(k + khalf) * HID;
            const float* br1 = br0 + HID;
            #pragma unroll
            for (int j = 0; j < 4; ++j) {
                const int col = n0 + j * 16 + l16;
                v2f bb; bb.x = br0[col]; bb.y = br1[col];
                acc[j] = wmma_f32(a, bb, acc[j]);
            }
        }
        #pragma unroll
        for (int j = 0; j < 4; ++j)
            #pragma unroll
            for (int rr = 0; rr < 8; ++rr)
                acc[j][rr] = 1.0f / (1.0f + __expf(-acc[j][rr]));

        __syncthreads();                       // all waves done reading old h
        float* ht = hs + (long)t * BATCH * HID;
        #pragma unroll
        for (int j = 0; j < 4; ++j) {
            const int col = n0 + j * 16 + l16;
            #pragma unroll
            for (int rr = 0; rr < 8; ++rr) {
                const float v = acc[j][rr];
                hbuf[(rowbase + rr) * HID + col]       = v;
                ht[(long)(rowbase + rr) * HID + col]   = v;
                if (t == STEPS - 1)
                    fin[(long)(rowbase + rr) * HID + col] = v;
            }
        }
        __syncthreads();
    }
}

// ---------------------------------------------------------------------------
// Phase 3: out = hs @ U + b2   (M=4096, N=50257, K=256) -- the 105 GFLOP GEMM.
// grid (32, ceil(V/128)), block 256 (8 waves): block tile 128(M) x 128(N),
// wave tile 32(M) x 64(N) -> 8 accumulators (2 A tiles x 4 B tiles).
// The 128-col U tile (shared by all 4 M-waves) is staged in LDS via
// global_load_async_to_lds_b128, double-buffered (2 x 32K x 128 floats),
// synchronized with s_wait_asynccnt + workgroup barriers so the async DMA
// of chunk c+1 overlaps the 8 WMMA k-steps of chunk c.
// ---------------------------------------------------------------------------
#define UCH_ROWS 32                      // K rows per staged chunk
#define UCH_ELEMS (UCH_ROWS * 128)       // floats per chunk (16 KB)

__global__ __launch_bounds__(256) void rnn_vocab(
    const float* __restrict__ hs,    // [STEPS*BATCH][HID]
    const float* __restrict__ U,     // [HID][VOCAB]
    const float* __restrict__ b2,    // [VOCAB]
    float*       __restrict__ out)   // [STEPS*BATCH][VOCAB]
{
    __shared__ float ubuf[2 * UCH_ELEMS];            // 32 KB double buffer

    const int lane  = threadIdx.x & 31;
    const int wid   = threadIdx.x >> 5;              // 0..7
    const int wm    = wid >> 1;                      // 0..3
    const int wn    = wid & 1;                       // 0..1
    const int m0    = blockIdx.x * 128 + wm * 32;
    const int nblk  = blockIdx.y * 128;
    const int nloc  = wn * 64;
    const int l16   = lane & 15;
    const int khalf = (lane >> 4) * 2;

    // ---- async staging setup: each wave DMAs 4 rows (512 B each) per chunk
    // lane covers 4 consecutive floats; clamp keeps b128 in-bounds+16B-aligned
    int gcol = nblk + 4 * lane;
    const int gclamp = (VOCAB - 4) & ~3;
    if (gcol > gclamp) gcol = gclamp;

    // ---- accumulators: bias-initialized; edge columns clamped (EXEC all-ones)
    v8f acc0[4], acc1[4];
    #pragma unroll
    for (int j = 0; j < 4; ++j) {
        const int c = nblk + nloc + j * 16 + l16;
        const float bias = (c < VOCAB) ? b2[c] : 0.0f;
        #pragma unroll
        for (int rr = 0; rr < 8; ++rr) { acc0[j][rr] = bias; acc1[j][rr] = bias; }
    }

    const float* a0base = hs + (long)(m0 + l16) * HID + khalf;        // A rows 0..15
    const float* a1base = a0base + 16 * HID;                          // A rows 16..31

    // issue chunk 0 DMA
    #pragma unroll
    for (int r2 = 0; r2 < 4; ++r2) {
        const int row = wid * 4 + r2;
        const uint64_t ga = (uint64_t)(uintptr_t)(U + (long)row * VOCAB + gcol);
        const uint32_t la = (uint32_t)(uintptr_t)(ubuf + row * 128 + 4 * lane);
        asm volatile("global_load_async_to_lds_b128 %0, %1, off"
                     :: "v"(la), "v"(ga) : "memory");
    }

    const int NCHUNK = HID / UCH_ROWS;               // 8
    for (int c = 0; c < NCHUNK; ++c) {
        const float* cur = ubuf + (c & 1) * UCH_ELEMS;
        if (c + 1 < NCHUNK) {
            float* nxt = ubuf + ((c + 1) & 1) * UCH_ELEMS;
            #pragma unroll
            for (int r2 = 0; r2 < 4; ++r2) {
                const int row = wid * 4 + r2;
                const uint64_t ga = (uint64_t)(uintptr_t)
                    (U + (long)((c + 1) * UCH_ROWS + row) * VOCAB + gcol);
                const uint32_t la = (uint32_t)(uintptr_t)(nxt + row * 128 + 4 * lane);
                asm volatile("global_load_async_to_lds_b128 %0, %1, off"
                             :: "v"(la), "v"(ga) : "memory");
            }
            asm volatile("s_wait_asynccnt 0x4" ::: "memory");  // chunk c done, c+1 in flight
        } else {
            asm volatile("s_wait_asynccnt 0x0" ::: "memory");
        }
        __syncthreads();                              // chunk c visible to all waves

        const float* a0p = a0base + c * UCH_ROWS;
        const float* a1p = a1base + c * UCH_ROWS;
        #pragma unroll
        for (int kk = 0; kk < UCH_ROWS; kk += 4) {
            v2f a0 = *(const v2f*)(a0p + kk);
            v2f a1 = *(const v2f*)(a1p + kk);
            const float* brow = cur + (kk + khalf) * 128 + nloc + l16;   // LDS
            #pragma unroll
            for (int j = 0; j < 4; ++j) {
                v2f bb; bb.x = brow[j * 16]; bb.y = brow[j * 16 + 128];
                acc0[j] = wmma_f32(a0, bb, acc0[j]);
                acc1[j] = wmma_f32(a1, bb, acc1[j]);
            }
        }
        __syncthreads();                              // done reading cur before refill
    }

    const int rowbase = m0 + (lane >> 4) * 8;
    #pragma unroll
    for (int j = 0; j < 4; ++j) {
        const int c = nblk + nloc + j * 16 + l16;
        if (c < VOCAB) {
            float* p0 = out + (long)rowbase * VOCAB + c;
            float* p1 = p0 + (long)16 * VOCAB;
            #pragma unroll
            for (int rr = 0; rr < 8; ++rr) {
                p0[(long)rr * VOCAB] = acc0[j][rr];
                p1[(long)rr * VOCAB] = acc1[j][rr];
            }
        }
    }
}

// ---------------------------------------------------------------------------
extern "C" void kernel_launch(void* const* d_in, const int* in_sizes, int n_in,
                              void* d_out, int out_size, void* d_ws, size_t ws_size,
                              hipStream_t stream) {
    const int*   input_x = (const int*)  d_in[0];   // [BATCH][STEPS]
    const float* h0      = (const float*)d_in[1];   // [BATCH][HID]
    const float* emb     = (const float*)d_in[2];   // [VOCAB][EMBED]
    const float* Hm      = (const float*)d_in[3];   // [HID][HID]
    const float* Im      = (const float*)d_in[4];   // [EMBED][HID]
    const float* b1      = (const float*)d_in[5];   // [HID]
    const float* U       = (const float*)d_in[6];   // [HID][VOCAB]
    const float* b2      = (const float*)d_in[7];   // [VOCAB]

    float* out = (float*)d_out;                               // logits then final state
    float* fin = out + (size_t)STEPS * BATCH * (size_t)VOCAB; // [BATCH][HID]

    float* xI = (float*)d_ws;                                 // 4 MB
    float* hs = xI + (size_t)STEPS * BATCH * HID;             // 4 MB

    rnn_xproj<<<dim3(64, 4),                    256, 0, stream>>>(input_x, emb, Im, xI);
    rnn_recur<<<1,                              512, 0, stream>>>(xI, Hm, b1, h0, hs, fin);
    rnn_vocab<<<dim3(32, (VOCAB + 127) / 128),  256, 0, stream>>>(hs, U, b2, out);
}